// LogicLayer_26938034881129
// MI455X (gfx1250) — compile-verified
//
#include <hip/hip_runtime.h>

#define D_FEAT   64
#define K_DIM    128
#define TILE_M   16
#define M_TILES  2          // 16-edge tiles per wave (share B fragments)
#define WAVES    4          // waves per block in the GEMM kernel
#define WSTRIDE  72         // LDS row stride (floats) for W tile  -> disjoint bank halves
#define FSTRIDE  132        // LDS row stride (floats) for feats   -> bank = 4*row+col

typedef __attribute__((ext_vector_type(2))) float v2f;
typedef __attribute__((ext_vector_type(4))) float v4f;
typedef __attribute__((ext_vector_type(8))) float v8f;

__device__ __forceinline__ v8f wmma_f32_16x16x4(v2f a, v2f b, v8f c) {
  // 8 args: (neg_a, A, neg_b, B, c_mod, C, reuse_a, reuse_b)
  return __builtin_amdgcn_wmma_f32_16x16x4_f32(false, a, false, b, (short)0, c,
                                               false, false);
}

// ---------------------------------------------------------------------------
// K0: zero output + counts, compute softmax(edge_attention) into ws[0..T)
// ---------------------------------------------------------------------------
__global__ void k0_init(float* __restrict__ out, float* __restrict__ wsf,
                        const float* __restrict__ attn, int N, int T) {
  size_t idx    = (size_t)blockIdx.x * blockDim.x + threadIdx.x;
  size_t stride = (size_t)gridDim.x * blockDim.x;
  size_t nOut = (size_t)N * D_FEAT;
  size_t nCnt = (size_t)T * N;
  for (size_t i = idx; i < nOut; i += stride) out[i] = 0.0f;
  for (size_t i = idx; i < nCnt; i += stride) wsf[8 + i] = 0.0f;
  if (idx == 0) {
    float m = attn[0];
    for (int t = 1; t < T; ++t) m = fmaxf(m, attn[t]);
    float e[8];
    float s = 0.0f;
    for (int t = 0; t < T && t < 8; ++t) { e[t] = __expf(attn[t] - m); s += e[t]; }
    for (int t = 0; t < T && t < 8; ++t) wsf[t] = e[t] / s;
  }
}

// ---------------------------------------------------------------------------
// K1: per-(t, src-node) edge counts
// ---------------------------------------------------------------------------
__global__ void k1_count(const int* __restrict__ edges, float* __restrict__ cnt,
                         int N, int E, int T) {
  size_t idx    = (size_t)blockIdx.x * blockDim.x + threadIdx.x;
  size_t stride = (size_t)gridDim.x * blockDim.x;
  size_t total  = (size_t)T * E;
  for (size_t i = idx; i < total; i += stride) {
    size_t t = i / (size_t)E;
    size_t e = i - t * (size_t)E;
    int s = edges[t * 2u * (size_t)E + e];           // edges[t][0][e]
    atomicAdd(&cnt[t * (size_t)N + s], 1.0f);
  }
}

// ---------------------------------------------------------------------------
// K2: counts -> folded scale = softmax_w[t] / max(count, 1)
// ---------------------------------------------------------------------------
__global__ void k2_scale(float* __restrict__ wsf, int N, int T) {
  size_t idx   = (size_t)blockIdx.x * blockDim.x + threadIdx.x;
  size_t total = (size_t)T * N;
  if (idx >= total) return;
  size_t t = idx / (size_t)N;
  float c = wsf[8 + idx];
  wsf[8 + idx] = wsf[t] / fmaxf(c, 1.0f);
}

// ---------------------------------------------------------------------------
// K3: per-wave 32-edge block (2 x 16-row tiles sharing B fragments):
//     gather feats -> LDS, WMMA f32 16x16x4 over K=128 (8 WMMA / K-chunk),
//     bias + relu + (w_t / count) scale, atomic scatter into out.
// ---------------------------------------------------------------------------
__global__ __launch_bounds__(WAVES * 32)
void k3_gemm_scatter(const float* __restrict__ x, const float* __restrict__ W,
                     const float* __restrict__ b, const int* __restrict__ edges,
                     const float* __restrict__ scale, float* __restrict__ out,
                     int N, int E) {
  extern __shared__ float lds[];
  float* sW = lds;                       // K_DIM * WSTRIDE floats
  float* sF = lds + K_DIM * WSTRIDE;     // WAVES * 32 * FSTRIDE floats

  const int t     = blockIdx.y;
  const int tid   = threadIdx.x;
  const int wave  = tid >> 5;
  const int lane  = tid & 31;
  const int row16 = lane & 15;
  const int half  = lane >> 4;

  // Stage W[t] (128x64) into LDS, row stride WSTRIDE.
  const float* Wt = W + (size_t)t * K_DIM * D_FEAT;
  for (int i = tid; i < K_DIM * (D_FEAT / 4); i += WAVES * 32) {
    int r  = i >> 4;            // 0..127
    int c4 = (i & 15) << 2;     // 0..60
    v4f v = *(const v4f*)(Wt + r * D_FEAT + c4);
    float* p = sW + r * WSTRIDE + c4;
    p[0] = v.x; p[1] = v.y; p[2] = v.z; p[3] = v.w;
  }
  __syncthreads();

  const int nblk32 = (E + M_TILES * TILE_M - 1) / (M_TILES * TILE_M);
  const int blk    = blockIdx.x * WAVES + wave;
  if (blk >= nblk32) return;

  const int* srcE = edges + (size_t)t * 2 * E;
  const int* dstE = srcE + E;
  float* sFw = sF + wave * (M_TILES * TILE_M * FSTRIDE);

  const int e0 = blk * (M_TILES * TILE_M);
  // Edge ids for the two tiles (lane&15 indexes the row within each tile).
  int ed0 = e0 + row16;
  int ed1 = e0 + TILE_M + row16;
  int cl0 = (ed0 < E) ? ed0 : (E - 1);
  int cl1 = (ed1 < E) ? ed1 : (E - 1);
  const int srcn0 = srcE[cl0];
  const int dstn0 = dstE[cl0];
  const int srcn1 = srcE[cl1];
  const int dstn1 = dstE[cl1];

  // Gather 32 feat rows ([x[src] | x[dst]], 128 f32) into LDS — coalesced
  // 256B row loads; each lane writes 8B per row segment.
  #pragma unroll 4
  for (int r = 0; r < TILE_M; ++r) {
    int sn0 = __shfl(srcn0, r, 32);
    int dn0 = __shfl(dstn0, r, 32);
    int sn1 = __shfl(srcn1, r, 32);
    int dn1 = __shfl(dstn1, r, 32);
    v2f vs0 = *(const v2f*)(x + (size_t)sn0 * D_FEAT + 2 * lane);
    v2f vd0 = *(const v2f*)(x + (size_t)dn0 * D_FEAT + 2 * lane);
    v2f vs1 = *(const v2f*)(x + (size_t)sn1 * D_FEAT + 2 * lane);
    v2f vd1 = *(const v2f*)(x + (size_t)dn1 * D_FEAT + 2 * lane);
    *(v2f*)(sFw + r * FSTRIDE + 2 * lane)                               = vs0;
    *(v2f*)(sFw + r * FSTRIDE + D_FEAT + 2 * lane)                      = vd0;
    *(v2f*)(sFw + (TILE_M + r) * FSTRIDE + 2 * lane)                    = vs1;
    *(v2f*)(sFw + (TILE_M + r) * FSTRIDE + D_FEAT + 2 * lane)           = vd1;
  }

  v8f accA0 = {}, accA1 = {}, accA2 = {}, accA3 = {};
  v8f accB0 = {}, accB1 = {}, accB2 = {}, accB3 = {};

  // A fragments (16x4 f32): lane half selects K pair {0,1} vs {2,3}.
  const float* aRow0 = sFw + row16 * FSTRIDE + 2 * half;
  const float* aRow1 = sFw + (TILE_M + row16) * FSTRIDE + 2 * half;

  #pragma unroll
  for (int c = 0; c < K_DIM / 4; ++c) {
    v2f a0 = *(const v2f*)(aRow0 + 4 * c);
    v2f a1 = *(const v2f*)(aRow1 + 4 * c);
    // B fragment (4x16 f32): low lanes = low K pair; VGPR j = K + j.
    const float* bRow0 = sW + (4 * c + 2 * half) * WSTRIDE + row16;
    const float* bRow1 = bRow0 + WSTRIDE;
    v2f b0; b0.x = bRow0[0];  b0.y = bRow1[0];
    v2f b1; b1.x = bRow0[16]; b1.y = bRow1[16];
    v2f b2; b2.x = bRow0[32]; b2.y = bRow1[32];
    v2f b3; b3.x = bRow0[48]; b3.y = bRow1[48];
    accA0 = wmma_f32_16x16x4(a0, b0, accA0);
    accB0 = wmma_f32_16x16x4(a1, b0, accB0);
    accA1 = wmma_f32_16x16x4(a0, b1, accA1);
    accB1 = wmma_f32_16x16x4(a1, b1, accB1);
    accA2 = wmma_f32_16x16x4(a0, b2, accA2);
    accB2 = wmma_f32_16x16x4(a1, b2, accB2);
    accA3 = wmma_f32_16x16x4(a0, b3, accA3);
    accB3 = wmma_f32_16x16x4(a1, b3, accB3);
  }

  // Epilogue: C/D layout — VGPR v, M = v + 8*half, N = lane&15 (per 16-col tile).
  const float* bt = b + (size_t)t * D_FEAT;
  float bias0 = bt[0  + row16];
  float bias1 = bt[16 + row16];
  float bias2 = bt[32 + row16];
  float bias3 = bt[48 + row16];

  v8f accsA[4] = {accA0, accA1, accA2, accA3};
  v8f accsB[4] = {accB0, accB1, accB2, accB3};
  float biases[4] = {bias0, bias1, bias2, bias3};

  #pragma unroll
  for (int v = 0; v < 8; ++v) {
    int m = v + 8 * half;
    // Tile A (edges e0 + m)
    {
      int  sn    = __shfl(srcn0, m, 32);
      bool valid = (e0 + m) < E;
      float sc   = scale[(size_t)t * N + sn];
      #pragma unroll
      for (int nt = 0; nt < 4; ++nt) {
        float val = accsA[nt][v] + biases[nt];
        val = fmaxf(val, 0.0f) * sc;
        if (valid)
          atomicAdd(out + (size_t)sn * D_FEAT + nt * 16 + row16, val);
      }
    }
    // Tile B (edges e0 + 16 + m)
    {
      int  sn    = __shfl(srcn1, m, 32);
      bool valid = (e0 + TILE_M + m) < E;
      float sc   = scale[(size_t)t * N + sn];
      #pragma unroll
      for (int nt = 0; nt < 4; ++nt) {
        float val = accsB[nt][v] + biases[nt];
        val = fmaxf(val, 0.0f) * sc;
        if (valid)
          atomicAdd(out + (size_t)sn * D_FEAT + nt * 16 + row16, val);
      }
    }
  }
}

// ---------------------------------------------------------------------------
extern "C" void kernel_launch(void* const* d_in, const int* in_sizes, int n_in,
                              void* d_out, int out_size, void* d_ws, size_t ws_size,
                              hipStream_t stream) {
  const float* x     = (const float*)d_in[0];   // (N, 64)
  const float* W     = (const float*)d_in[1];   // (T, 128, 64)
  const float* b     = (const float*)d_in[2];   // (T, 64)
  const float* attn  = (const float*)d_in[3];   // (T,)
  const int*   edges = (const int*)d_in[4];     // (T, 2, E) int32
  float* out = (float*)d_out;                   // (N, 64)
  float* wsf = (float*)d_ws;                    // [0..8): softmax w, [8..): T*N scale

  const int N = in_sizes[0] / D_FEAT;
  const int T = in_sizes[3];
  const int E = in_sizes[4] / (2 * T);

  k0_init<<<2048, 256, 0, stream>>>(out, wsf, attn, N, T);
  k1_count<<<2048, 256, 0, stream>>>(edges, wsf + 8, N, E, T);
  {
    size_t total = (size_t)T * N;
    k2_scale<<<(unsigned)((total + 255) / 256), 256, 0, stream>>>(wsf, N, T);
  }
  const int nblk32 = (E + M_TILES * TILE_M - 1) / (M_TILES * TILE_M);
  dim3 grid((nblk32 + WAVES - 1) / WAVES, T);
  size_t ldsBytes = (size_t)(K_DIM * WSTRIDE + WAVES * M_TILES * TILE_M * FSTRIDE)
                    * sizeof(float);
  k3_gemm_scatter<<<grid, WAVES * 32, ldsBytes, stream>>>(x, W, b, edges,
                                                          wsf + 8, out, N, E);
}